// TensorProductFlowModel_7928509628866
// MI455X (gfx1250) — compile-verified
//
#include <hip/hip_runtime.h>
#include <math.h>

typedef __attribute__((ext_vector_type(2))) float v2f;
typedef __attribute__((ext_vector_type(8))) float v8f;

#define SQRT3      1.7320508075688772f
#define INV_S16    0.25f          /* 1/sqrt(16) */
#define INV_S4     0.5f           /* 1/sqrt(4)  */
#define INV_S12    0.28867513459481287f /* 1/sqrt(3*4) */
#define INV_S8     0.35355339059327373f /* 1/sqrt(2*4) */

// ---------------------------------------------------------------- utilities
__global__ void zero_fill(float* __restrict__ p, int n) {
    int i = blockIdx.x * blockDim.x + threadIdx.x;
    if (i < n) p[i] = 0.f;
}

__global__ void prep_edges(const float* __restrict__ pos,
                           const int* __restrict__ eidx,
                           float* __restrict__ sh, float* __restrict__ counts,
                           int E) {
    int e = blockIdx.x * blockDim.x + threadIdx.x;
    if (e >= E) return;
    int s = eidx[e], d = eidx[E + e];
    float vx = pos[d*3+0] - pos[s*3+0];
    float vy = pos[d*3+1] - pos[s*3+1];
    float vz = pos[d*3+2] - pos[s*3+2];
    float inv = SQRT3 / (sqrtf(vx*vx + vy*vy + vz*vz) + 1e-8f);
    sh[e*3+0] = vx*inv; sh[e*3+1] = vy*inv; sh[e*3+2] = vz*inv;
    unsafeAtomicAdd(&counts[d], 1.f);
}

__global__ void make_deg_inv(const float* __restrict__ counts,
                             float* __restrict__ dg, int N) {
    int i = blockIdx.x * blockDim.x + threadIdx.x;
    if (i < N) dg[i] = 1.f / fmaxf(counts[i], 1.f);
}

__global__ void finalize_k(const float* __restrict__ acc,
                           const float* __restrict__ dg,
                           const float* __restrict__ res,
                           float* __restrict__ out, int N, int W) {
    int i = blockIdx.x * blockDim.x + threadIdx.x;
    if (i >= N * W) return;
    float v = acc[i] * dg[i / W];
    if (res) v += res[i];
    out[i] = v;
}

// --------------------------------------------------- fused per-layer kernel
// One wave (32 threads) handles a tile of 16 edges.
template <int L>
__global__ __launch_bounds__(32) void edge_tp_kernel(
    const float* __restrict__ eattr,
    const float* __restrict__ f0,     // [N,16] scalar feats (src of gather)
    const float* __restrict__ f1o,    // [N,4,3] or null
    const float* __restrict__ f1e,    // [N,4,3] or null
    const int*   __restrict__ eidx,   // [2,E]
    const float* __restrict__ sh1,    // [E,3]
    const float* __restrict__ w1, const float* __restrict__ b1,  // [48,48],[48]
    const float* __restrict__ w2, const float* __restrict__ b2,  // [48,WN],[WN]
    float* __restrict__ acc0, float* __restrict__ acc1o, float* __restrict__ acc1e,
    int E)
{
    constexpr int WN  = (L == 0) ? 320 : (L == 1) ? 416 : 24;
    constexpr int PAD = (L == 2) ? 32 : WN;
    constexpr int NT2 = PAD / 16;

    __shared__ float s_ein[16][48];
    __shared__ float s_h[16][48];
    __shared__ float s_w[16][PAD];
    __shared__ float s_sh[16][3];
    __shared__ float s_v[16][12];
    __shared__ float s_u[16][12];
    __shared__ int   s_src[16];
    __shared__ int   s_dst[16];

    const int lane = threadIdx.x;
    const int half = lane >> 4;
    const int l16  = lane & 15;
    const int e0   = blockIdx.x * 16;

    // ---- stage per-edge metadata
    if (lane < 16) {
        int e = e0 + lane;
        bool ok = e < E;
        s_src[lane] = ok ? eidx[e] : 0;
        s_dst[lane] = ok ? eidx[E + e] : 0;
        #pragma unroll
        for (int k = 0; k < 3; ++k) s_sh[lane][k] = ok ? sh1[e*3 + k] : 0.f;
    }
    __syncthreads();

    // ---- gather MLP input tile: [edge_attr | s[src] | s[dst]]
    for (int idx = lane; idx < 256; idx += 32) {
        int m = idx >> 4, c = idx & 15;
        bool ok = (e0 + m) < E;
        int e = e0 + m;
        s_ein[m][c]      = ok ? eattr[e*16 + c]        : 0.f;
        s_ein[m][16 + c] = ok ? f0[s_src[m]*16 + c]    : 0.f;
        s_ein[m][32 + c] = ok ? f0[s_dst[m]*16 + c]    : 0.f;
    }
    if constexpr (L >= 1) {
        for (int idx = lane; idx < 192; idx += 32) {
            int m = idx / 12, c = idx % 12;
            bool ok = (e0 + m) < E;
            s_v[m][c] = ok ? f1o[s_src[m]*12 + c] : 0.f;
            if constexpr (L == 2) s_u[m][c] = ok ? f1e[s_src[m]*12 + c] : 0.f;
        }
    }
    __syncthreads();

    // ---- GEMM1: h = relu(ein @ w1 + b1)   [16x48]·[48x48]
    v2f af[12];
    #pragma unroll
    for (int kt = 0; kt < 12; ++kt) {
        int kk = kt*4 + half*2;
        v2f a; a[0] = s_ein[l16][kk]; a[1] = s_ein[l16][kk + 1];
        af[kt] = a;
    }
    #pragma unroll
    for (int nt = 0; nt < 3; ++nt) {
        v8f acc = {};
        int col = nt*16 + l16;
        #pragma unroll
        for (int kt = 0; kt < 12; ++kt) {
            int kk = kt*4 + half*2;
            v2f b; b[0] = w1[kk*48 + col]; b[1] = w1[(kk + 1)*48 + col];
            acc = __builtin_amdgcn_wmma_f32_16x16x4_f32(
                      false, af[kt], false, b, (short)0, acc, false, false);
        }
        float bv = b1[col];
        #pragma unroll
        for (int j = 0; j < 8; ++j)
            s_h[j + half*8][col] = fmaxf(acc[j] + bv, 0.f);
    }
    __syncthreads();

    // ---- GEMM2: w = h @ w2 + b2   [16x48]·[48xWN]
    #pragma unroll
    for (int kt = 0; kt < 12; ++kt) {
        int kk = kt*4 + half*2;
        v2f a; a[0] = s_h[l16][kk]; a[1] = s_h[l16][kk + 1];
        af[kt] = a;
    }
    for (int nt = 0; nt < NT2; ++nt) {
        v8f acc = {};
        int col = nt*16 + l16;
        bool okc = col < WN;
        #pragma unroll
        for (int kt = 0; kt < 12; ++kt) {
            int kk = kt*4 + half*2;
            v2f b;
            b[0] = okc ? w2[kk*WN + col]       : 0.f;
            b[1] = okc ? w2[(kk + 1)*WN + col] : 0.f;
            acc = __builtin_amdgcn_wmma_f32_16x16x4_f32(
                      false, af[kt], false, b, (short)0, acc, false, false);
        }
        float bv = okc ? b2[col] : 0.f;
        #pragma unroll
        for (int j = 0; j < 8; ++j)
            s_w[j + half*8][col] = acc[j] + bv;
    }
    __syncthreads();

    // ---- tensor product + atomic scatter: lane m owns edge m
    if (lane < 16 && (e0 + lane) < E) {
        const int m = lane;
        const float* wr = s_w[m];
        const float h0 = s_sh[m][0], h1 = s_sh[m][1], h2 = s_sh[m][2];
        const int d = s_dst[m];
        float x[16];
        #pragma unroll
        for (int i = 0; i < 16; ++i) x[i] = s_ein[m][16 + i];

        if constexpr (L == 0) {
            for (int o = 0; o < 16; ++o) {                       // 0e x Y0 -> 0e
                float t = 0.f;
                for (int i = 0; i < 16; ++i) t += x[i] * wr[i*16 + o];
                unsafeAtomicAdd(&acc0[d*16 + o], t * INV_S16);
            }
            for (int o = 0; o < 4; ++o) {                        // 0e x Y1 -> 1o
                float t = 0.f;
                for (int i = 0; i < 16; ++i) t += x[i] * wr[256 + i*4 + o];
                t *= INV_S16;
                unsafeAtomicAdd(&acc1o[d*12 + o*3 + 0], t * h0);
                unsafeAtomicAdd(&acc1o[d*12 + o*3 + 1], t * h1);
                unsafeAtomicAdd(&acc1o[d*12 + o*3 + 2], t * h2);
            }
        } else if constexpr (L == 1) {
            float v[4][3];
            #pragma unroll
            for (int i = 0; i < 4; ++i)
                for (int k = 0; k < 3; ++k) v[i][k] = s_v[m][i*3 + k];
            float msg0[16], m1o[4][3], m1e[4][3];
            for (int o = 0; o < 16; ++o) {                       // 0e x Y0 -> 0e
                float t = 0.f;
                for (int i = 0; i < 16; ++i) t += x[i] * wr[i*16 + o];
                msg0[o] = t * INV_S16;
            }
            for (int o = 0; o < 4; ++o) {                        // 0e x Y1 -> 1o
                float t = 0.f;
                for (int i = 0; i < 16; ++i) t += x[i] * wr[256 + i*4 + o];
                t *= INV_S16;
                m1o[o][0] = t * h0; m1o[o][1] = t * h1; m1o[o][2] = t * h2;
            }
            for (int o = 0; o < 4; ++o)                          // 1o x Y0 -> 1o
                for (int k = 0; k < 3; ++k) {
                    float t = 0.f;
                    for (int i = 0; i < 4; ++i) t += v[i][k] * wr[320 + i*4 + o];
                    m1o[o][k] += t * INV_S4;
                }
            float dt[4];                                          // 1o . Y1 -> 0e
            for (int i = 0; i < 4; ++i)
                dt[i] = v[i][0]*h0 + v[i][1]*h1 + v[i][2]*h2;
            for (int o = 0; o < 16; ++o) {
                float t = 0.f;
                for (int i = 0; i < 4; ++i) t += dt[i] * wr[336 + i*16 + o];
                msg0[o] += t * INV_S12;
            }
            float cr[4][3];                                       // 1o x Y1 -> 1e
            for (int i = 0; i < 4; ++i) {
                cr[i][0] = v[i][1]*h2 - v[i][2]*h1;
                cr[i][1] = v[i][2]*h0 - v[i][0]*h2;
                cr[i][2] = v[i][0]*h1 - v[i][1]*h0;
            }
            for (int o = 0; o < 4; ++o)
                for (int k = 0; k < 3; ++k) {
                    float t = 0.f;
                    for (int i = 0; i < 4; ++i) t += cr[i][k] * wr[400 + i*4 + o];
                    m1e[o][k] = t * INV_S8;
                }
            for (int o = 0; o < 16; ++o)
                unsafeAtomicAdd(&acc0[d*16 + o], msg0[o]);
            for (int o = 0; o < 4; ++o)
                for (int k = 0; k < 3; ++k) {
                    unsafeAtomicAdd(&acc1o[d*12 + o*3 + k], m1o[o][k]);
                    unsafeAtomicAdd(&acc1e[d*12 + o*3 + k], m1e[o][k]);
                }
        } else {  // L == 2 : out = 1x1o
            float v[4][3], u[4][3];
            #pragma unroll
            for (int i = 0; i < 4; ++i)
                for (int k = 0; k < 3; ++k) {
                    v[i][k] = s_v[m][i*3 + k];
                    u[i][k] = s_u[m][i*3 + k];
                }
            float msg[3];
            float t = 0.f;                                        // 0e x Y1 -> 1o
            for (int i = 0; i < 16; ++i) t += x[i] * wr[i];
            t *= INV_S16;
            msg[0] = t * h0; msg[1] = t * h1; msg[2] = t * h2;
            for (int k = 0; k < 3; ++k) {                         // 1o x Y0 -> 1o
                float s = 0.f;
                for (int i = 0; i < 4; ++i) s += v[i][k] * wr[16 + i];
                msg[k] += s * INV_S4;
            }
            float cr[4][3];                                       // 1e x Y1 -> 1o
            for (int i = 0; i < 4; ++i) {
                cr[i][0] = u[i][1]*h2 - u[i][2]*h1;
                cr[i][1] = u[i][2]*h0 - u[i][0]*h2;
                cr[i][2] = u[i][0]*h1 - u[i][1]*h0;
            }
            for (int k = 0; k < 3; ++k) {
                float s = 0.f;
                for (int i = 0; i < 4; ++i) s += cr[i][k] * wr[20 + i];
                msg[k] += s * INV_S8;
            }
            for (int k = 0; k < 3; ++k)
                unsafeAtomicAdd(&acc1o[d*3 + k], msg[k]);
        }
    }
}

// ------------------------------------------------------------------- driver
extern "C" void kernel_launch(void* const* d_in, const int* in_sizes, int n_in,
                              void* d_out, int out_size, void* d_ws, size_t ws_size,
                              hipStream_t stream) {
    const float* node0e = (const float*)d_in[0];
    const float* pos    = (const float*)d_in[1];
    const float* eattr  = (const float*)d_in[2];
    const float* W1[3], *B1[3], *W2[3], *B2[3];
    for (int l = 0; l < 3; ++l) {
        W1[l] = (const float*)d_in[3 + l*4 + 0];
        B1[l] = (const float*)d_in[3 + l*4 + 1];
        W2[l] = (const float*)d_in[3 + l*4 + 2];
        B2[l] = (const float*)d_in[3 + l*4 + 3];
    }
    const int* eidx = (const int*)d_in[15];
    const int N = in_sizes[0] / 16;
    const int E = in_sizes[15] / 2;

    float* p = (float*)d_ws;
    auto take = [&](size_t n) { float* r = p; p += n; return r; };
    float* sh1    = take((size_t)E * 3);
    float* counts = take(N);
    float* dg     = take(N);
    float* b0     = take((size_t)N * 16);
    float* b1o    = take((size_t)N * 12);
    float* a0     = take((size_t)N * 16);
    float* a1o    = take((size_t)N * 12);
    float* a1e    = take((size_t)N * 12);
    float* out3   = take((size_t)N * 3);
    float* acc0   = take((size_t)N * 16);
    float* acc1o  = take((size_t)N * 12);
    float* acc1e  = take((size_t)N * 12);

    const int T = 256;
    auto cdiv = [](int a, int b) { return (a + b - 1) / b; };
    auto zero = [&](float* q, int n) {
        zero_fill<<<cdiv(n, T), T, 0, stream>>>(q, n);
    };

    // prep: sh1, degrees
    zero(counts, N);
    prep_edges<<<cdiv(E, T), T, 0, stream>>>(pos, eidx, sh1, counts, E);
    make_deg_inv<<<cdiv(N, T), T, 0, stream>>>(counts, dg, N);

    const int EG = cdiv(E, 16);

    // ---- layer 0: {0e:16} -> {0e:16, 1o:4}
    zero(acc0, N*16); zero(acc1o, N*12);
    edge_tp_kernel<0><<<EG, 32, 0, stream>>>(
        eattr, node0e, nullptr, nullptr, eidx, sh1,
        W1[0], B1[0], W2[0], B2[0], acc0, acc1o, nullptr, E);
    finalize_k<<<cdiv(N*16, T), T, 0, stream>>>(acc0, dg, node0e, b0, N, 16);
    finalize_k<<<cdiv(N*12, T), T, 0, stream>>>(acc1o, dg, nullptr, b1o, N, 12);

    // ---- layer 1: {0e:16,1o:4} -> {0e:16, 1o:4, 1e:4}
    zero(acc0, N*16); zero(acc1o, N*12); zero(acc1e, N*12);
    edge_tp_kernel<1><<<EG, 32, 0, stream>>>(
        eattr, b0, b1o, nullptr, eidx, sh1,
        W1[1], B1[1], W2[1], B2[1], acc0, acc1o, acc1e, E);
    finalize_k<<<cdiv(N*16, T), T, 0, stream>>>(acc0, dg, b0, a0, N, 16);
    finalize_k<<<cdiv(N*12, T), T, 0, stream>>>(acc1o, dg, b1o, a1o, N, 12);
    finalize_k<<<cdiv(N*12, T), T, 0, stream>>>(acc1e, dg, nullptr, a1e, N, 12);

    // ---- layer 2: {0e:16,1o:4,1e:4} -> {1o:1}
    zero(out3, N*3);
    edge_tp_kernel<2><<<EG, 32, 0, stream>>>(
        eattr, a0, a1o, a1e, eidx, sh1,
        W1[2], B1[2], W2[2], B2[2], nullptr, out3, nullptr, E);
    finalize_k<<<cdiv(N*3, T), T, 0, stream>>>(out3, dg, nullptr,
                                               (float*)d_out, N, 3);
}